// LinkPredictorGNN_22376779612381
// MI455X (gfx1250) — compile-verified
//
#include <hip/hip_runtime.h>
#include <hip/hip_bf16.h>

typedef __attribute__((ext_vector_type(2))) float v2f;
typedef __attribute__((ext_vector_type(8))) float v8f;

#define HIDDEN 128
#define TILES_O (HIDDEN / 16)   // 8 feature tiles
#define XS_STRIDE 132           // 128 + 4 pad -> conflict-free A-fragment reads

// ---------------- degree / normalization ----------------

__global__ __launch_bounds__(256) void deg_init_kernel(int* __restrict__ deg, int N) {
    int n = blockIdx.x * blockDim.x + threadIdx.x;
    if (n < N) deg[n] = 1;  // self loop contributes 1
}

__global__ __launch_bounds__(256) void deg_accum_kernel(const int* __restrict__ dst,
                                                        int* __restrict__ deg, int E) {
    int e = blockIdx.x * blockDim.x + threadIdx.x;
    if (e < E) atomicAdd(&deg[dst[e]], 1);
}

__global__ __launch_bounds__(256) void dinv_kernel(const int* __restrict__ deg,
                                                   float* __restrict__ dinv, int N) {
    int n = blockIdx.x * blockDim.x + threadIdx.x;
    if (n < N) dinv[n] = rsqrtf((float)deg[n]);  // deg >= 1 always
}

// ---------------- h = x @ W^T via V_WMMA_F32_16X16X4_F32 ----------------
// Block = 8 waves, one node-tile (16 rows of x) per block; wave w computes the
// 16x16 tile for feature-tile 'w'. The shared 16x128 x-tile (one contiguous
// 8KB span) is DMA'd into LDS with CDNA5 async global->LDS loads (ASYNCcnt),
// then each wave reads A fragments from LDS and B fragments (W, L2-resident)
// from global.

__global__ __launch_bounds__(256) void gemm_wmma_kernel(const float* __restrict__ x,
                                                        const float* __restrict__ W,
                                                        float* __restrict__ h,
                                                        int n_tiles_m) {
    __shared__ float xs[16 * XS_STRIDE];

    const int tm   = blockIdx.x;            // node tile (grid sized exactly)
    const int wave = threadIdx.x >> 5;      // feature tile 'to' (0..7), wave32
    const int lane = threadIdx.x & 31;
    const int t    = threadIdx.x;

    // ---- async-stage x tile: 16 rows x 32 float4 chunks = 512 b128 transfers,
    //      2 per thread. LDS rows padded to XS_STRIDE floats.
    const unsigned lds_base = (unsigned)(size_t)(&xs[0]);
#pragma unroll
    for (int c = t; c < 512; c += 256) {
        const int row  = c >> 5;            // /32
        const int col4 = c & 31;
        const unsigned lds_off = lds_base + (unsigned)(row * XS_STRIDE + col4 * 4) * 4u;
        const unsigned long long ga =
            (unsigned long long)(size_t)(x + ((long)(tm * 16 + row) * HIDDEN + col4 * 4));
        asm volatile("global_load_async_to_lds_b128 %0, %1, off"
                     :: "v"(lds_off), "v"(ga) : "memory");
    }
    asm volatile("s_wait_asynccnt 0x0" ::: "memory");
    __syncthreads();

    // ---- WMMA main loop
    const int half = lane >> 4;             // 0: lanes 0-15, 1: lanes 16-31
    const int l15  = lane & 15;
    const int koff = half * 2;              // K pair select per ISA 32-bit layout

    const float* __restrict__ arow = &xs[l15 * XS_STRIDE];
    const long rowB = (long)(wave * 16 + l15) * HIDDEN;  // W row = output feature

    v8f c = {0.f, 0.f, 0.f, 0.f, 0.f, 0.f, 0.f, 0.f};

#pragma unroll
    for (int k0 = 0; k0 < HIDDEN; k0 += 4) {
        // A fragment (16x4 f32) from LDS: VGPR0=K(k0+koff), VGPR1=K(k0+koff+1)
        v2f a;
        a.x = arow[k0 + koff];
        a.y = arow[k0 + koff + 1];
        // B fragment (4x16 f32): B[k][n] = W[n][k], same K striping
        v2f b;
        b.x = W[rowB + k0 + koff];
        b.y = W[rowB + k0 + koff + 1];
        c = __builtin_amdgcn_wmma_f32_16x16x4_f32(
                /*neg_a=*/false, a, /*neg_b=*/false, b,
                /*c_mod=*/(short)0, c, /*reuse_a=*/false, /*reuse_b=*/false);
    }

    // D layout: VGPR v holds M = half*8 + v, N = l15
#pragma unroll
    for (int v = 0; v < 8; ++v) {
        const long row = (long)(tm * 16 + half * 8 + v);
        h[row * HIDDEN + wave * 16 + l15] = c[v];
    }
}

// ---------------- out = b + self-loop term ----------------

__global__ __launch_bounds__(256) void out_init_kernel(const float* __restrict__ h,
                                                       const float* __restrict__ dinv,
                                                       const float* __restrict__ bias,
                                                       float* __restrict__ out,
                                                       long total) {
    long i = (long)blockIdx.x * blockDim.x + threadIdx.x;
    if (i >= total) return;
    long n = i >> 7;          // /128
    int  f = (int)(i & 127);
    float di = dinv[n];
    out[i] = bias[f] + h[i] * di * di;
}

// ---------------- edge gather-scale-scatter ----------------
// One wave per edge; each lane owns 4 consecutive features (coalesced float4
// gather from h, 4x global_atomic_add_f32 scatter into out). h/out are
// L2-resident (51.2MB each vs 192MB L2).

__global__ __launch_bounds__(256) void edge_agg_kernel(const int* __restrict__ src,
                                                       const int* __restrict__ dst,
                                                       const float* __restrict__ h,
                                                       const float* __restrict__ dinv,
                                                       float* __restrict__ out,
                                                       int E) {
    const int wave = threadIdx.x >> 5;
    const int lane = threadIdx.x & 31;
    const int e = blockIdx.x * 8 + wave;
    if (e >= E) return;
    const int s = src[e];
    const int d = dst[e];
    const float nrm = dinv[s] * dinv[d];

    const float4* hs = (const float4*)(h + (long)s * HIDDEN);
    float4 v = hs[lane];

    float* od = out + (long)d * HIDDEN + lane * 4;
    atomicAdd(od + 0, v.x * nrm);
    atomicAdd(od + 1, v.y * nrm);
    atomicAdd(od + 2, v.z * nrm);
    atomicAdd(od + 3, v.w * nrm);
}

// ---------------- launch ----------------

extern "C" void kernel_launch(void* const* d_in, const int* in_sizes, int n_in,
                              void* d_out, int out_size, void* d_ws, size_t ws_size,
                              hipStream_t stream) {
    const float* x    = (const float*)d_in[0];
    const int*   ei   = (const int*)d_in[1];   // [2, E] flattened: src then dst
    const float* W    = (const float*)d_in[2];
    const float* bias = (const float*)d_in[3];
    float* out = (float*)d_out;

    const int N = in_sizes[0] / HIDDEN;        // 100000
    const int E = in_sizes[1] / 2;             // 1600000
    const int* src = ei;
    const int* dst = ei + E;

    // workspace partition: h [N*128 f32] | deg [N i32] | dinv [N f32]
    float* h    = (float*)d_ws;
    int*   deg  = (int*)((char*)d_ws + (size_t)N * HIDDEN * sizeof(float));
    float* dinv = (float*)((char*)deg + (size_t)N * sizeof(int));

    const int nt_nodes = (N + 255) / 256;
    const int nt_edges = (E + 255) / 256;

    // 1) degrees (incl. self loops), 2) dinv
    deg_init_kernel<<<nt_nodes, 256, 0, stream>>>(deg, N);
    deg_accum_kernel<<<nt_edges, 256, 0, stream>>>(dst, deg, E);
    dinv_kernel<<<nt_nodes, 256, 0, stream>>>(deg, dinv, N);

    // 3) h = x @ W^T  (one node-tile per block, 8 feature-tile waves)
    const int n_tiles_m = N / 16;              // 6250, exact
    gemm_wmma_kernel<<<n_tiles_m, 256, 0, stream>>>(x, W, h, n_tiles_m);

    // 4) out = bias + self-loop contribution
    const long total = (long)N * HIDDEN;
    out_init_kernel<<<(int)((total + 255) / 256), 256, 0, stream>>>(h, dinv, bias, out, total);

    // 5) edge scatter: one wave per edge
    edge_agg_kernel<<<(E + 7) / 8, 256, 0, stream>>>(src, dst, h, dinv, out, E);
}